// CopyGenerator_33285996544702
// MI455X (gfx1250) — compile-verified
//
#include <hip/hip_runtime.h>

typedef __attribute__((ext_vector_type(2))) float v2f;
typedef __attribute__((ext_vector_type(8))) float v8f;

#define N_ROWS 2048
#define D_DIM  1024
#define V_DIM  50257
#define S_LEN  400
#define BATCH  32
#define T_LEN  (N_ROWS / BATCH)   // 64
#define CVOCAB 600

// ---------------------------------------------------------------------------
// Kernel 1: p_copy[row] = sigmoid(dot(hidden[row,:], W) + b)
// GEMV via V_WMMA_F32_16X16X4_F32: one wave handles 16 rows, K marched in 4s.
// A-tile: lane L -> row (L&15), K-pair ((L>>4)*2). B-tile: all 16 columns
// replicate W, so per-lane B uses the same K indexing as A.
// ---------------------------------------------------------------------------
__global__ void __launch_bounds__(32) pcopy_wmma_kernel(
    const float* __restrict__ hidden, const float* __restrict__ W,
    const float* __restrict__ b, float* __restrict__ pcopy)
{
    const int lane = threadIdx.x;            // 0..31, one wave per block
    const int m    = lane & 15;
    const int kq   = (lane >> 4) << 1;       // 0 or 2
    const int row  = blockIdx.x * 16 + m;
    const float* __restrict__ hrow = hidden + (size_t)row * D_DIM;

    v8f c = {};
    #pragma unroll 4
    for (int k = 0; k < D_DIM; k += 4) {
        v2f a, bb;
        a.x  = hrow[k + kq];
        a.y  = hrow[k + kq + 1];
        bb.x = W[k + kq];
        bb.y = W[k + kq + 1];
        // D = A(16x4,f32) * B(4x16,f32) + C ; emits v_wmma_f32_16x16x4_f32
        c = __builtin_amdgcn_wmma_f32_16x16x4_f32(
                /*neg_a=*/false, a, /*neg_b=*/false, bb,
                /*c_mod=*/(short)0, c, /*reuse_a=*/false, /*reuse_b=*/false);
    }

    // D layout (f32 16x16): lanes 0-15 hold M=0..7 in c[0..7]; lanes 16-31 hold
    // M=8..15. Every column is identical, so column-0 lanes write the results.
    if ((lane & 15) == 0) {
        const float bias  = b[0];
        const int   rbase = blockIdx.x * 16 + ((lane >> 4) << 3);
        #pragma unroll
        for (int i = 0; i < 8; ++i) {
            float z = c[i] + bias;
            pcopy[rbase + i] = 1.0f / (1.0f + __expf(-z));
        }
    }
}

// ---------------------------------------------------------------------------
// Kernel 2: out[row, :] = softmax(logits[row, :]) * (1 - p_copy[row])
// One 1024-thread (32-wave) workgroup per row. Online softmax pass + scaled
// write pass; second read of the 201 KB row should hit L2.
// ---------------------------------------------------------------------------
__global__ void __launch_bounds__(1024) softmax_scale_kernel(
    const float* __restrict__ logits, const float* __restrict__ pcopy,
    float* __restrict__ out)
{
    const int row = blockIdx.x;
    const int tid = threadIdx.x;
    const int nt  = blockDim.x;
    const float* __restrict__ x = logits + (size_t)row * V_DIM;
    float* __restrict__ y       = out    + (size_t)row * V_DIM;

    // Pass 1: per-thread online max/sum
    float m = -__builtin_inff();
    float s = 0.0f;
    for (int i = tid; i < V_DIM; i += nt) {
        float v  = x[i];
        float nm = fmaxf(m, v);
        s = s * __expf(m - nm) + __expf(v - nm);
        m = nm;
    }

    // Wave32 reduction via lane shuffles
    #pragma unroll
    for (int off = 16; off > 0; off >>= 1) {
        float mo = __shfl_xor(m, off, 32);
        float so = __shfl_xor(s, off, 32);
        float nm = fmaxf(m, mo);
        s = s * __expf(m - nm) + so * __expf(mo - nm);
        m = nm;
    }

    // Cross-wave reduction through LDS
    __shared__ float sm[32], ss[32];
    __shared__ float bm, binv;
    const int wid = tid >> 5, lid = tid & 31;
    if (lid == 0) { sm[wid] = m; ss[wid] = s; }
    __syncthreads();
    if (tid == 0) {
        float M = sm[0], S = ss[0];
        const int nw = nt >> 5;
        for (int w = 1; w < nw; ++w) {
            float mo = sm[w], so = ss[w];
            float nm = fmaxf(M, mo);
            S = S * __expf(M - nm) + so * __expf(mo - nm);
            M = nm;
        }
        bm   = M;
        binv = (1.0f - pcopy[row]) / S;
    }
    __syncthreads();

    // Pass 2: scaled exp write (row re-read served by L2)
    const float M = bm, scale = binv;
    for (int i = tid; i < V_DIM; i += nt) {
        y[i] = __expf(x[i] - M) * scale;
    }
}

// ---------------------------------------------------------------------------
// Kernel 3: copy_prob[row, :] = scatter-add of attn[row, s] * p_copy[row]
// into bins src_map[row / T_LEN, s]. LDS accumulation with ds_add_f32 atomics
// (duplicate indices accumulate, matching the reference .at[].add semantics).
// ---------------------------------------------------------------------------
__global__ void __launch_bounds__(128) copy_scatter_kernel(
    const float* __restrict__ attn, const int* __restrict__ src_map,
    const float* __restrict__ pcopy, float* __restrict__ cprob)
{
    __shared__ float acc[CVOCAB];
    const int row = blockIdx.x;
    const int tid = threadIdx.x;

    for (int j = tid; j < CVOCAB; j += blockDim.x) acc[j] = 0.0f;
    __syncthreads();

    const int   b    = row / T_LEN;
    const float pc   = pcopy[row];
    const float* __restrict__ arow = attn    + (size_t)row * S_LEN;
    const int*   __restrict__ srow = src_map + (size_t)b   * S_LEN;

    for (int s = tid; s < S_LEN; s += blockDim.x) {
        atomicAdd(&acc[srow[s]], arow[s] * pc);   // ds_add_f32
    }
    __syncthreads();

    float* __restrict__ crow = cprob + (size_t)row * CVOCAB;
    for (int j = tid; j < CVOCAB; j += blockDim.x) crow[j] = acc[j];
}

// ---------------------------------------------------------------------------
extern "C" void kernel_launch(void* const* d_in, const int* in_sizes, int n_in,
                              void* d_out, int out_size, void* d_ws, size_t ws_size,
                              hipStream_t stream)
{
    const float* hidden = (const float*)d_in[0];   // (N, D)
    const float* orig   = (const float*)d_in[1];   // (N, V)
    const float* attn   = (const float*)d_in[2];   // (N, S)
    const int*   smap   = (const int*)  d_in[3];   // (B, S)
    const float* W      = (const float*)d_in[4];   // (1, D)
    const float* bcopy  = (const float*)d_in[5];   // (1,)

    float* out   = (float*)d_out;                  // out_prob (N*V) ++ copy_prob (N*CVOCAB)
    float* pcopy = (float*)d_ws;                   // scratch: N floats

    pcopy_wmma_kernel   <<<N_ROWS / 16, 32,   0, stream>>>(hidden, W, bcopy, pcopy);
    softmax_scale_kernel<<<N_ROWS,      1024, 0, stream>>>(orig, pcopy, out);
    copy_scatter_kernel <<<N_ROWS,      128,  0, stream>>>(attn, smap, pcopy,
                                                           out + (size_t)N_ROWS * V_DIM);
}